// DenSaE_4681514352805
// MI455X (gfx1250) — compile-verified
//
#include <hip/hip_runtime.h>
#include <cmath>
#include <cstdint>

// ---------------------------------------------------------------------------
// DenSaE / FISTA convolutional sparse coding for MI455X (gfx1250, wave32).
// Encoder conv (C_in=1 -> 64 ch): implicit GEMM on V_WMMA_F32_16X16X4_F32.
// Decoder convT (64 ch -> 1): GEMV on VALU with LDS tiling, staged by the
// Tensor Data Mover (tensor_load_to_lds + s_wait_tensorcnt), double-buffered.
// State loop fully on-device; FISTA momentum fused into the WMMA epilogue.
// ---------------------------------------------------------------------------

typedef __attribute__((ext_vector_type(2))) float v2f;
typedef __attribute__((ext_vector_type(8))) float v8f;
typedef __attribute__((ext_vector_type(4))) unsigned int u32x4;
typedef __attribute__((ext_vector_type(4))) int i32x4;
typedef __attribute__((ext_vector_type(8))) int i32x8;

constexpr int    BSZ = 32, HI = 128, WI = 128, HC = 122, WC = 122;
constexpr size_t S  = (size_t)BSZ * 32 * HC * WC;   // one 32-channel map set
constexpr size_t R  = (size_t)BSZ * HI * WI;        // image-sized map
constexpr float  LREC  = 1.0f / 100.0f;             // 1/L
constexpr float  THR   = 0.1f / 100.0f;             // lam/L
constexpr float  SCALE = 2.0f;                      // 1 + 1/rho

// ------------------------------------------------------------------ TDM issue
// Build a D# (ISA 8.3-8.6) for a 3-D tile load: tile0 x tile1 x tile2 elements
// of 4-byte data, row stride td0s, plane stride td1s (all in elements).
// tile2 == 0 makes it effectively 2-D. Must be called wave-uniformly.
__device__ inline void tdm_load_tile(uint32_t lds_off, const void* gptr,
                                     uint32_t tile0, uint32_t tile1, uint32_t tile2,
                                     uint32_t td0, uint64_t td0s,
                                     uint32_t td1, uint64_t td1s, uint32_t td2) {
  uint64_t ga = (uint64_t)(uintptr_t)gptr;
  u32x4 g0;
  g0[0] = 1u;                                           // count=1, user mode
  g0[1] = lds_off;                                      // LDS byte address
  g0[2] = (uint32_t)(ga & 0xFFFFFFFFu);                 // global_addr[31:0]
  g0[3] = (uint32_t)((ga >> 32) & 0x01FFFFFFu) | (2u << 30);  // [56:32] | type=2
  i32x8 g1;
  g1[0] = (int)(2u << 16);                              // data_size = 4B
  g1[1] = (int)((td0 & 0xFFFFu) << 16);                 // tensor_dim0[15:0]
  g1[2] = (int)(((td0 >> 16) & 0xFFFFu) | ((td1 & 0xFFFFu) << 16));
  g1[3] = (int)(((td1 >> 16) & 0xFFFFu) | ((tile0 & 0xFFFFu) << 16));
  g1[4] = (int)((tile1 & 0xFFFFu) | ((tile2 & 0xFFFFu) << 16));
  g1[5] = (int)(uint32_t)(td0s & 0xFFFFFFFFu);          // dim0 stride (y step)
  g1[6] = (int)((((uint32_t)(td0s >> 32)) & 0xFFFFu) |
                (((uint32_t)td1s & 0xFFFFu) << 16));    // dim1 stride (z step) lo
  g1[7] = (int)(uint32_t)((td1s >> 16) & 0xFFFFFFFFu);  // dim1 stride hi
  i32x4 g2;
  g2[0] = (int)td2;                                     // tensor_dim2
  g2[1] = 0;
  g2[2] = 0;                                            // dim2 stride (unused, 3-D)
  g2[3] = 0;
  i32x4 g3 = {0, 0, 0, 0};
#if defined(__clang_major__) && __clang_major__ >= 23
  i32x8 g4 = {0, 0, 0, 0, 0, 0, 0, 0};
  __builtin_amdgcn_tensor_load_to_lds(g0, g1, g2, g3, g4, 0);
#else
  __builtin_amdgcn_tensor_load_to_lds(g0, g1, g2, g3, 0);
#endif
}

// -------------------------------------------------- init: zero state buffers
__global__ void zero2_kernel(float* __restrict__ a, float* __restrict__ b, size_t n) {
  size_t i = (size_t)blockIdx.x * blockDim.x + threadIdx.x;
  size_t stride = (size_t)gridDim.x * blockDim.x;
  for (; i < n; i += stride) { a[i] = 0.0f; b[i] = 0.0f; }
}

// ------------------------------------------- pack weights (fold 1/L, scale, pad)
__global__ void pack_weights_kernel(const float* __restrict__ A, const float* __restrict__ B,
                                    float* __restrict__ Wg, float* __restrict__ Wr,
                                    float* __restrict__ Wf) {
  int idx = blockIdx.x * blockDim.x + threadIdx.x;
  if (idx < 64 * 64) {
    int oc = idx >> 6, k = idx & 63;
    float w = 0.0f;
    if (k < 49) w = (oc < 32) ? A[oc * 49 + k] : B[(oc - 32) * 49 + k];
    Wg[idx] = w * LREC;
  }
  if (idx < 64 * 49) {
    int oc = idx / 49, k = idx % 49;
    float w = (oc < 32) ? A[oc * 49 + k] : B[(oc - 32) * 49 + k];
    Wr[idx] = (oc < 32) ? SCALE * w : w;
    Wf[idx] = w;
  }
}

// -------------------------------------------------------- decoder (convT, GEMV)
// mode 0: out0 = y - (Ax~ + Bu~); mode 1: out0 = Ax+Bu, out1 = Ax, out2 = Bu.
// Grid (8,8,32): 16x16 output tile, 256 threads. Channels in chunks of 8.
// Interior tiles: TDM DMA (22x22x8 tile) into a double-buffered LDS patch,
// next chunk's DMA overlapped with the current chunk's 392-FMA reduction.
__global__ __launch_bounds__(256)
void convT_kernel(const float* __restrict__ srcx, const float* __restrict__ srcu,
                  const float* __restrict__ W, const float* __restrict__ y,
                  float* __restrict__ out0, float* __restrict__ out1,
                  float* __restrict__ out2, int mode) {
  constexpr int CHUNKF = 8 * 22 * 22;   // 3872 floats per chunk
  __shared__ float sw[64 * 49];
  __shared__ float spd[2 * CHUNKF];

  const int tid = threadIdx.x;
  const int ti = tid >> 4, tj = tid & 15;
  const int j0 = blockIdx.x * 16, i0 = blockIdx.y * 16, b = blockIdx.z;

  for (int idx = tid; idx < 64 * 49; idx += 256) sw[idx] = W[idx];

  float accA = 0.0f, accB = 0.0f;
  auto accum_chunk = [&](int chunk, const float* sc) {
    float partial = 0.0f;
#pragma unroll
    for (int cc = 0; cc < 8; ++cc) {
      const float* wrow = &sw[(chunk * 8 + cc) * 49];
      const float* prow = &sc[cc * 484];
#pragma unroll
      for (int a = 0; a < 7; ++a)
#pragma unroll
        for (int bb = 0; bb < 7; ++bb)
          partial = fmaf(prow[(ti + 6 - a) * 22 + (tj + 6 - bb)], wrow[a * 7 + bb], partial);
    }
    if (chunk < 4) accA += partial; else accB += partial;
  };
  auto issue_chunk = [&](int c) {   // wave-uniform TDM issue, 8 channel planes
    const float* src = (c < 4) ? srcx : srcu;
    int ch0 = (c & 3) * 8;
    const float* gp = &src[(((size_t)b * 32 + ch0) * HC + (i0 - 6)) * WC + (j0 - 6)];
    tdm_load_tile((uint32_t)(uintptr_t)(&spd[(c & 1) * CHUNKF]), gp,
                  /*tile*/22, 22, 8,
                  /*tensor*/(uint32_t)WC, (uint64_t)WC,
                  (uint32_t)HC, (uint64_t)HC * WC, 32u);
  };

  const bool interior = (i0 >= 6) && (i0 + 15 < HC) && (j0 >= 6) && (j0 + 15 < HC);
  if (interior) {
    if (tid < 32) issue_chunk(0);
    for (int chunk = 0; chunk < 8; ++chunk) {
      if (tid < 32) {
        if (chunk + 1 < 8) {
          issue_chunk(chunk + 1);                       // prefetch next chunk
          __builtin_amdgcn_s_wait_tensorcnt(1);         // current chunk landed
        } else {
          __builtin_amdgcn_s_wait_tensorcnt(0);
        }
      }
      __syncthreads();
      accum_chunk(chunk, &spd[(chunk & 1) * CHUNKF]);
      __syncthreads();
    }
  } else {
    for (int chunk = 0; chunk < 8; ++chunk) {
      for (int idx = tid; idx < CHUNKF; idx += 256) {   // scalar zero-fill stage
        int cc = idx / 484, rem = idx % 484;
        int rr = rem / 22, col = rem % 22;
        int gi = i0 + rr - 6, gj = j0 + col - 6;
        int ch = chunk * 8 + cc;
        const float* src = (ch < 32) ? srcx : srcu;
        float v = 0.0f;
        if (gi >= 0 && gi < HC && gj >= 0 && gj < WC)
          v = src[(((size_t)b * 32 + (ch & 31)) * HC + gi) * WC + gj];
        spd[idx] = v;
      }
      __syncthreads();
      accum_chunk(chunk, &spd[0]);
      __syncthreads();
    }
  }

  const size_t p = ((size_t)b * HI + (i0 + ti)) * WI + (j0 + tj);
  if (mode == 0) {
    out0[p] = y[p] - (accA + accB);
  } else {
    out0[p] = accA + accB;
    out1[p] = accA;
    out2[p] = accB;
  }
}

// ------------------------------------------ encoder (conv) via f32 WMMA + fuse
// Grid (122,32), 128 threads = 4 waves; wave w owns channels [16w,16w+16).
// Residual patch (7 rows x 24 cols) staged by TDM (scalar fallback for the
// right edge tile); row 7 kept zero so K=49..51 pad contributes nothing.
__global__ __launch_bounds__(128)
void grad_wmma_kernel(const float* __restrict__ res, const float* __restrict__ Wg,
                      float* __restrict__ x_tmp, float* __restrict__ u_tmp,
                      float* __restrict__ x_new, float* __restrict__ u_new,
                      float cmom, float thr) {
  __shared__ float sres[8 * 24];

  const int tid  = threadIdx.x;
  const int lane = tid & 31;
  const int wave = tid >> 5;
  const int i = blockIdx.x;           // output row, 0..121
  const int b = blockIdx.y;           // batch

  const int m    = lane & 15;
  const int ksel = (lane < 16) ? 0 : 2;   // f32 16x16x4 operand layout: K 0/1 vs 2/3
  const int ch   = wave * 16 + m;

  v2f afrag[13];                      // A operand: weights for all 13 K-steps
#pragma unroll
  for (int t = 0; t < 13; ++t) {
    int kk = t * 4 + ksel;
    afrag[t].x = Wg[ch * 64 + kk];
    afrag[t].y = Wg[ch * 64 + kk + 1];
  }

  if (tid < 24) sres[7 * 24 + tid] = 0.0f;   // permanent zero pad row

  const int wv = __builtin_amdgcn_readfirstlane(wave);   // scalarize x/u split

  for (int jt = 0; jt < 8; ++jt) {
    const int j0 = jt * 16;
    if (jt < 7) {                     // cols j0..j0+23 <= 119: TDM-safe
      if (tid < 32) {
        tdm_load_tile((uint32_t)(uintptr_t)sres,
                      &res[((size_t)b * HI + i) * WI + j0],
                      /*tile*/24, 7, 0,
                      /*tensor*/(uint32_t)WI, (uint64_t)WI,
                      (uint32_t)HI, (uint64_t)HI * WI, 1u);
        __builtin_amdgcn_s_wait_tensorcnt(0);
      }
    } else {                          // right edge: zero-fill cols >= 128
      for (int idx = tid; idx < 7 * 24; idx += 128) {
        int rr = idx / 24, cc = idx % 24;
        int gc = j0 + cc;
        sres[idx] = (gc < WI) ? res[((size_t)b * HI + (i + rr)) * WI + gc] : 0.0f;
      }
    }
    if (jt + 1 < 8)
      __builtin_prefetch(&res[((size_t)b * HI + i) * WI + (jt + 1) * 16], 0, 1);
    __syncthreads();

    v8f acc = {};
    const int n = lane & 15;
#pragma unroll
    for (int t = 0; t < 13; ++t) {
      int k0 = t * 4 + ksel;
      int k1 = k0 + 1;
      v2f bfrag;
      bfrag.x = sres[(k0 / 7) * 24 + n + (k0 % 7)];
      bfrag.y = sres[(k1 / 7) * 24 + n + (k1 % 7)];
      acc = __builtin_amdgcn_wmma_f32_16x16x4_f32(
          false, afrag[t], false, bfrag, (short)0, acc, false, false);
    }

    const int j = j0 + n;             // D: VGPR r -> (M = r or r+8, N = lane%16)
    if (wv < 2) {                     // waves 0,1: x channels (scalar branch)
      if (j < WC) {
#pragma unroll
        for (int r = 0; r < 8; ++r) {
          int ch2 = wv * 16 + r + ((lane < 16) ? 0 : 8);
          size_t idx = (((size_t)b * 32 + ch2) * HC + i) * WC + j;
          float xn = x_tmp[idx] + acc[r];
          float xo = x_new[idx];                  // holds x_old
          x_new[idx] = xn;
          x_tmp[idx] = xn + cmom * (xn - xo);
        }
      }
    } else {                          // waves 2,3: u channels + shrinkage
      if (j < WC) {
#pragma unroll
        for (int r = 0; r < 8; ++r) {
          int ch2 = (wv - 2) * 16 + r + ((lane < 16) ? 0 : 8);
          size_t idx = (((size_t)b * 32 + ch2) * HC + i) * WC + j;
          float un = u_tmp[idx] + acc[r];
          float s  = fabsf(un) - thr;
          un = (s > 0.0f) ? copysignf(s, un) : 0.0f;
          float uo = u_new[idx];
          u_new[idx] = un;
          u_tmp[idx] = un + cmom * (un - uo);
        }
      }
    }
    __syncthreads();
  }
}

// ---------------------------------------------------------------------------
extern "C" void kernel_launch(void* const* d_in, const int* in_sizes, int n_in,
                              void* d_out, int out_size, void* d_ws, size_t ws_size,
                              hipStream_t stream) {
  (void)in_sizes; (void)n_in; (void)out_size; (void)ws_size;
  const float* y  = (const float*)d_in[0];
  const float* A  = (const float*)d_in[1];
  const float* Bw = (const float*)d_in[2];
  float* out = (float*)d_out;
  float* ws  = (float*)d_ws;

  // workspace: x_tmp | u_tmp | res | Wg(64x64) | Wr(64x49) | Wf(64x49)  (~124 MB)
  float* x_tmp = ws;
  float* u_tmp = ws + S;
  float* res   = ws + 2 * S;
  float* Wg    = ws + 2 * S + R;
  float* Wr    = Wg + 64 * 64;
  float* Wf    = Wr + 64 * 49;

  // d_out: y_hat | x_new | u_new | Ax_hat | Bu_hat  (x_new/u_new double as state)
  float* y_hat = out;
  float* x_new = out + R;
  float* u_new = out + R + S;
  float* Ax    = out + R + 2 * S;
  float* Bu    = out + 2 * R + 2 * S;

  zero2_kernel<<<4096, 256, 0, stream>>>(x_tmp, x_new, 2 * S);
  pack_weights_kernel<<<(64 * 64 + 255) / 256, 256, 0, stream>>>(A, Bw, Wg, Wr, Wf);

  float t = 1.0f;
  for (int it = 0; it < 15; ++it) {
    float tn = (1.0f + sqrtf(1.0f + 4.0f * t * t)) * 0.5f;
    float c  = (t - 1.0f) / tn;
    t = tn;
    convT_kernel<<<dim3(8, 8, 32), 256, 0, stream>>>(x_tmp, u_tmp, Wr, y,
                                                     res, nullptr, nullptr, 0);
    grad_wmma_kernel<<<dim3(122, 32), 128, 0, stream>>>(res, Wg, x_tmp, u_tmp,
                                                        x_new, u_new, c, THR);
  }
  convT_kernel<<<dim3(8, 8, 32), 256, 0, stream>>>(x_new, u_new, Wf, nullptr,
                                                   y_hat, Ax, Bu, 1);
}